// SessionGraph_82910048682598
// MI455X (gfx1250) — compile-verified
//
#include <hip/hip_runtime.h>
#include <math.h>

// ---------------- problem constants (match reference) ----------------
#define DIM       128
#define NUM_NODE  43097
#define BSESS     4096
#define LITEMS    50
#define NITEMS    (BSESS * LITEMS)   // 204800
#define EINT      (NITEMS * 4)       // 819200
#define NOUTCOLS  (NUM_NODE - 1)     // 43096
#define ALPHA_NEG 0.2f

typedef float v2f __attribute__((ext_vector_type(2)));
typedef float v8f __attribute__((ext_vector_type(8)));

// CDNA5 fp32 WMMA: D(16x16,f32) = A(16x4,f32) * B(4x16,f32) + C
__device__ __forceinline__ v8f wmma_f32_16x16x4(v2f a, v2f b, v8f c) {
    return __builtin_amdgcn_wmma_f32_16x16x4_f32(false, a, false, b, (short)0, c,
                                                 false, false);
}

// order-preserving float<->uint key for atomicMax-based segment max
__device__ __forceinline__ unsigned fkey(float f) {
    unsigned u = __float_as_uint(f);
    return (u & 0x80000000u) ? ~u : (u | 0x80000000u);
}
__device__ __forceinline__ float funkey(unsigned k) {
    unsigned u = (k & 0x80000000u) ? (k & 0x7FFFFFFFu) : ~k;
    return __uint_as_float(u);
}

__device__ __forceinline__ float wave_reduce_add(float v) {
    v += __shfl_xor(v, 16, 32);
    v += __shfl_xor(v, 8, 32);
    v += __shfl_xor(v, 4, 32);
    v += __shfl_xor(v, 2, 32);
    v += __shfl_xor(v, 1, 32);
    return v;
}

// ---------------- stage 1: edge softmax over 'interacts' ----------------
__global__ void edge_logits_kernel(const float* __restrict__ emb,
                                   const float* __restrict__ pi_w,
                                   const int* __restrict__ iid,
                                   const int* __restrict__ esrc,
                                   const int* __restrict__ edst,
                                   float* __restrict__ logits,
                                   unsigned* __restrict__ mxkey) {
    int wave = (blockIdx.x * blockDim.x + threadIdx.x) >> 5;
    int lane = threadIdx.x & 31;
    if (wave >= EINT) return;
    int s = esrc[wave], d = edst[wave];
    const float* hs = emb + (size_t)iid[s] * DIM;
    const float* hd = emb + (size_t)iid[d] * DIM;
    float acc = 0.f;
#pragma unroll
    for (int t = 0; t < 4; ++t) {
        int k = lane + 32 * t;
        acc += hs[k] * pi_w[k] * hd[k];
    }
    acc = wave_reduce_add(acc);
    if (lane == 0) {
        float lg = acc >= 0.f ? acc : ALPHA_NEG * acc;
        logits[wave] = lg;
        atomicMax(mxkey + d, fkey(lg));
    }
}

__global__ void edge_exp_kernel(const int* __restrict__ edst,
                                const unsigned* __restrict__ mxkey,
                                float* __restrict__ logits,
                                float* __restrict__ denom) {
    int e = blockIdx.x * blockDim.x + threadIdx.x;
    if (e >= EINT) return;
    int d = edst[e];
    float ex = expf(logits[e] - funkey(mxkey[d]));
    logits[e] = ex;
    atomicAdd(denom + d, ex);
}

__global__ void edge_accum_kernel(const float* __restrict__ emb,
                                  const int* __restrict__ iid,
                                  const int* __restrict__ esrc,
                                  const int* __restrict__ edst,
                                  const float* __restrict__ ex,
                                  const float* __restrict__ denom,
                                  float* __restrict__ ft) {
    int wave = (blockIdx.x * blockDim.x + threadIdx.x) >> 5;
    int lane = threadIdx.x & 31;
    if (wave >= EINT) return;
    int s = esrc[wave], d = edst[wave];
    float a = ex[wave] / denom[d];
    const float* hs = emb + (size_t)iid[s] * DIM;
    float* fd = ft + (size_t)d * DIM;
#pragma unroll
    for (int t = 0; t < 4; ++t) {
        int k = lane + 32 * t;
        atomicAdd(fd + k, hs[k] * a);
    }
}

// ---------------- stage 2: session aggregation ----------------
__global__ void mean_kernel(const float* __restrict__ ft,
                            const int* __restrict__ agg_src,
                            const int* __restrict__ agg_dst,
                            float* __restrict__ mean_raw,
                            float* __restrict__ cnt) {
    int wave = (blockIdx.x * blockDim.x + threadIdx.x) >> 5;
    int lane = threadIdx.x & 31;
    if (wave >= NITEMS) return;
    int b = agg_dst[wave];
    const float* src = ft + (size_t)agg_src[wave] * DIM;
    float* dst = mean_raw + (size_t)b * DIM;
#pragma unroll
    for (int t = 0; t < 4; ++t) {
        int k = lane + 32 * t;
        atomicAdd(dst + k, src[k]);
    }
    if (lane == 0) atomicAdd(cnt + b, 1.0f);
}

__global__ void f_kernel(const float* __restrict__ target_emb,
                         const float* __restrict__ r_w,
                         const float* __restrict__ mean_raw,
                         const float* __restrict__ cnt,
                         const int* __restrict__ tid,
                         float* __restrict__ fmat) {
    int t = blockIdx.x * blockDim.x + threadIdx.x;
    if (t >= BSESS * DIM) return;
    int b = t >> 7, j = t & 127;
    float invc = 1.0f / fmaxf(cnt[b], 1.0f);
    const float* rw = r_w + (size_t)j * (2 * DIM);
    const float* ht = target_emb + (size_t)tid[b] * DIM;
    const float* mn = mean_raw + (size_t)b * DIM;
    float acc = 0.f;
#pragma unroll 8
    for (int k = 0; k < DIM; ++k) acc += rw[k] * ht[k];
#pragma unroll 8
    for (int k = 0; k < DIM; ++k) acc += rw[DIM + k] * (mn[k] * invc);
    fmat[t] = acc;
}

// ---------------- coef via WMMA ----------------
// coef[i] = sum_j tanh( [ft[agg_src[i]] | pos_emb[pid[i]]] . q_w[j,:] ) * f[agg_dst[i], j]
// Straight-line: one wave owns a 16-item tile with ALL 8 j-tiles accumulating
// simultaneously (c[8] = 64 VGPRs). Per K-step: one A float2 shared across the
// 8 tiles + a batch of 8 independent B loads (immediate offsets off one base),
// then 8 independent WMMA chains -> deep load pipelining, no loop-invariant A
// hoisting, no spills.
__global__ void __launch_bounds__(128)
coef_wmma_kernel(const float* __restrict__ ft,
                 const float* __restrict__ pos_emb,
                 const float* __restrict__ q_w,
                 const float* __restrict__ fmat,
                 const int* __restrict__ pid,
                 const int* __restrict__ agg_src,
                 const int* __restrict__ agg_dst,
                 float* __restrict__ coef) {
    int wave = (blockIdx.x * blockDim.x + threadIdx.x) >> 5;
    int lane = threadIdx.x & 31;
    int half = lane >> 4, l16 = lane & 15;
    int m0 = wave * 16;                       // exact grid: m0+15 < NITEMS
    int i = m0 + l16;                         // A-matrix row this lane feeds
    const float* ftrow = ft + (size_t)agg_src[i] * DIM + 2 * half;
    const float* prow = pos_emb + (size_t)pid[i] * DIM + 2 * half;
    // B row for tile jt, column l16: q_w[jt*16 + l16, :]; tiles are 16384B apart
    const float* qbase = q_w + (size_t)l16 * (2 * DIM) + 2 * half;

    v8f c[8];
#pragma unroll
    for (int jt = 0; jt < 8; ++jt) c[jt] = (v8f){0.f,0.f,0.f,0.f,0.f,0.f,0.f,0.f};

#pragma unroll
    for (int k0 = 0; k0 < 128; k0 += 4) {     // K-part 1: ft half of X
        v2f a = *(const v2f*)(ftrow + k0);
        v2f b[8];
#pragma unroll
        for (int jt = 0; jt < 8; ++jt)
            b[jt] = *(const v2f*)(qbase + (size_t)jt * 16 * (2 * DIM) + k0);
#pragma unroll
        for (int jt = 0; jt < 8; ++jt) c[jt] = wmma_f32_16x16x4(a, b[jt], c[jt]);
    }
#pragma unroll
    for (int k0 = 128; k0 < 256; k0 += 4) {   // K-part 2: pos_emb half of X
        v2f a = *(const v2f*)(prow + (k0 - 128));
        v2f b[8];
#pragma unroll
        for (int jt = 0; jt < 8; ++jt)
            b[jt] = *(const v2f*)(qbase + (size_t)jt * 16 * (2 * DIM) + k0);
#pragma unroll
        for (int jt = 0; jt < 8; ++jt) c[jt] = wmma_f32_16x16x4(a, b[jt], c[jt]);
    }

    // epilogue: fragment (row = m0 + 8*half + v, col = jt*16 + l16)
    int sessv[8];
#pragma unroll
    for (int v = 0; v < 8; ++v) sessv[v] = agg_dst[m0 + 8 * half + v];
    float rowacc[8] = {0.f, 0.f, 0.f, 0.f, 0.f, 0.f, 0.f, 0.f};
#pragma unroll
    for (int jt = 0; jt < 8; ++jt) {
        int j0 = jt * 16;
#pragma unroll
        for (int v = 0; v < 8; ++v) {
            float val = tanhf(c[jt][v]) * fmat[(size_t)sessv[v] * DIM + j0 + l16];
            val += __shfl_xor(val, 8, 32);    // reduce 16 cols (stays within half)
            val += __shfl_xor(val, 4, 32);
            val += __shfl_xor(val, 2, 32);
            val += __shfl_xor(val, 1, 32);
            rowacc[v] += val;
        }
    }
    if (l16 == 0) {
#pragma unroll
        for (int v = 0; v < 8; ++v) coef[m0 + 8 * half + v] = rowacc[v];
    }
}

// wave per item: select[sess] += ft[agg_src[i]] * coef[i]
__global__ void select_kernel(const float* __restrict__ ft,
                              const float* __restrict__ coef,
                              const int* __restrict__ agg_src,
                              const int* __restrict__ agg_dst,
                              float* __restrict__ sel) {
    int wave = (blockIdx.x * blockDim.x + threadIdx.x) >> 5;
    int lane = threadIdx.x & 31;
    if (wave >= NITEMS) return;
    float c = coef[wave];
    const float* src = ft + (size_t)agg_src[wave] * DIM;
    float* dst = sel + (size_t)agg_dst[wave] * DIM;
#pragma unroll
    for (int t = 0; t < 4; ++t) {
        int k = lane + 32 * t;
        atomicAdd(dst + k, src[k] * c);
    }
}

// ---------------- stage 3: scores = select @ emb[1:].T via fp32 WMMA ----------------
// Each wave owns one m-tile and NT=4 consecutive n-tiles: per K-step 1 A load
// shared by 4 tiles + 4 batched B loads + 4 independent WMMA chains.
// Output is write-once (706 MB) -> non-temporal stores keep L2 for emb/select.
#define NT 4
__global__ void __launch_bounds__(128)
scores_wmma_kernel(const float* __restrict__ sel,
                   const float* __restrict__ emb,
                   float* __restrict__ out) {
    int lane = threadIdx.x & 31;
    int wib = threadIdx.x >> 5;
    int mtile = blockIdx.y;
    int ntile0 = (blockIdx.x * 4 + wib) * NT;
    int half = lane >> 4, l16 = lane & 15;

    const float* ap = sel + (size_t)(mtile * 16 + l16) * DIM + 2 * half;
    const float* bp[NT];
#pragma unroll
    for (int j = 0; j < NT; ++j) {
        int ncol = (ntile0 + j) * 16 + l16;
        int r = (ncol < NOUTCOLS) ? (ncol + 1) : 0;   // clamp; stores guarded below
        bp[j] = emb + (size_t)r * DIM + 2 * half;
        __builtin_prefetch(bp[j], 0, 3);              // global_prefetch_b8
    }

    v8f c[NT];
#pragma unroll
    for (int j = 0; j < NT; ++j) c[j] = (v8f){0.f,0.f,0.f,0.f,0.f,0.f,0.f,0.f};

#pragma unroll
    for (int k0 = 0; k0 < 128; k0 += 4) {
        v2f a = *(const v2f*)(ap + k0);
        v2f b[NT];
#pragma unroll
        for (int j = 0; j < NT; ++j) b[j] = *(const v2f*)(bp[j] + k0);
#pragma unroll
        for (int j = 0; j < NT; ++j) c[j] = wmma_f32_16x16x4(a, b[j], c[j]);
    }

    // D layout: VGPR v -> row mtile*16 + 8*half + v, col = (ntile0+j)*16 + l16
    int crow0 = mtile * 16 + half * 8;
#pragma unroll
    for (int j = 0; j < NT; ++j) {
        int ncol = (ntile0 + j) * 16 + l16;
        if (ncol < NOUTCOLS) {
            float* o = out + (size_t)crow0 * NOUTCOLS + ncol;
#pragma unroll
            for (int v = 0; v < 8; ++v)
                __builtin_nontemporal_store(c[j][v], o + (size_t)v * NOUTCOLS);
        }
    }
}

// ---------------- host-side orchestration ----------------
extern "C" void kernel_launch(void* const* d_in, const int* in_sizes, int n_in,
                              void* d_out, int out_size, void* d_ws, size_t ws_size,
                              hipStream_t stream) {
    (void)in_sizes; (void)n_in; (void)out_size; (void)ws_size;
    const float* emb        = (const float*)d_in[0];
    const float* pos_emb    = (const float*)d_in[1];
    const float* target_emb = (const float*)d_in[2];
    const float* pi_w       = (const float*)d_in[3];
    const float* q_w        = (const float*)d_in[4];
    const float* r_w        = (const float*)d_in[5];
    const int*   iid        = (const int*)d_in[6];
    const int*   pid        = (const int*)d_in[7];
    const int*   tid        = (const int*)d_in[8];
    const int*   int_src    = (const int*)d_in[9];
    const int*   int_dst    = (const int*)d_in[10];
    const int*   agg_src    = (const int*)d_in[11];
    const int*   agg_dst    = (const int*)d_in[12];
    float* out = (float*)d_out;

    // workspace layout (floats); zero-initialized region first
    float* ws = (float*)d_ws;
    const size_t off_ft     = 0;                              // N*128
    const size_t off_mx     = off_ft + (size_t)NITEMS * DIM;  // N (uint keys)
    const size_t off_denom  = off_mx + NITEMS;                // N
    const size_t off_mean   = off_denom + NITEMS;             // B*128
    const size_t off_cnt    = off_mean + (size_t)BSESS * DIM; // B
    const size_t off_select = off_cnt + BSESS;                // B*128
    const size_t zero_end   = off_select + (size_t)BSESS * DIM;
    const size_t off_logits = zero_end;                       // E (fully overwritten)
    const size_t off_f      = off_logits + EINT;              // B*128 (overwritten)
    const size_t off_coef   = off_f + (size_t)BSESS * DIM;    // N (overwritten)

    float*    ft     = ws + off_ft;
    unsigned* mxkey  = (unsigned*)(ws + off_mx);
    float*    denom  = ws + off_denom;
    float*    meanr  = ws + off_mean;
    float*    cnt    = ws + off_cnt;
    float*    sel    = ws + off_select;
    float*    logits = ws + off_logits;
    float*    fmat   = ws + off_f;
    float*    coef   = ws + off_coef;

    hipMemsetAsync(d_ws, 0, zero_end * sizeof(float), stream);

    // stage 1: edge softmax + weighted aggregation
    edge_logits_kernel<<<EINT / 8, 256, 0, stream>>>(emb, pi_w, iid, int_src,
                                                     int_dst, logits, mxkey);
    edge_exp_kernel<<<EINT / 256, 256, 0, stream>>>(int_dst, mxkey, logits, denom);
    edge_accum_kernel<<<EINT / 8, 256, 0, stream>>>(emb, iid, int_src, int_dst,
                                                    logits, denom, ft);

    // stage 2: session mean, f, coef, select
    mean_kernel<<<NITEMS / 8, 256, 0, stream>>>(ft, agg_src, agg_dst, meanr, cnt);
    f_kernel<<<(BSESS * DIM) / 256, 256, 0, stream>>>(target_emb, r_w, meanr, cnt,
                                                      tid, fmat);
    coef_wmma_kernel<<<(NITEMS / 16) / 4, 128, 0, stream>>>(ft, pos_emb, q_w, fmat,
                                                            pid, agg_src, agg_dst,
                                                            coef);
    select_kernel<<<NITEMS / 8, 256, 0, stream>>>(ft, coef, agg_src, agg_dst, sel);

    // stage 3: scores GEMM, fp32 WMMA; block = 4 waves x 4 n-tiles = 16 n-tiles
    const int ntiles = (NOUTCOLS + 15) / 16;          // 2694
    dim3 grid((ntiles + 16 - 1) / 16, BSESS / 16);    // (169, 256)
    scores_wmma_kernel<<<grid, 128, 0, stream>>>(sel, emb, out);
}